// EntityClassify_88897233093156
// MI455X (gfx1250) — compile-verified
//
#include <hip/hip_runtime.h>

#define H 128
#define OUTD 16
#define NAx 100000
#define NBx 100000
#define NDx 50000
#define ASTR 132  // padded LDS row stride (floats) to avoid bank conflicts

typedef __attribute__((ext_vector_type(2))) float v2f;
typedef __attribute__((ext_vector_type(8))) float v8f;

// ---------------- utility kernels ----------------

__global__ void zero_kernel(float* __restrict__ p, size_t n) {
    size_t i = (size_t)blockIdx.x * blockDim.x + threadIdx.x;
    size_t stride = (size_t)gridDim.x * blockDim.x;
    for (; i < n; i += stride) p[i] = 0.0f;
}

// Wt[r][n][k] = coef[r,0]*basis[0][k][n] + coef[r,1]*basis[1][k][n]
// (stored K-contiguous so the GEMM B-fragment is a single b64 load per step)
__global__ void compute_Wt_kernel(const float* __restrict__ basis,
                                  const float* __restrict__ coef,
                                  float* __restrict__ Wt, int dout) {
    int idx = blockIdx.x * blockDim.x + threadIdx.x;
    int per = H * dout;
    if (idx >= 4 * per) return;
    int r = idx / per, rest = idx % per;
    int n = rest / H, k = rest % H;
    Wt[idx] = coef[r * 2 + 0] * basis[(size_t)k * dout + n] +
              coef[r * 2 + 1] * basis[(size_t)per + (size_t)k * dout + n];
}

// One wave32 per edge: gather src row (float4 per lane), fp32 atomics into agg[dst].
__global__ void scatter_kernel(const float* __restrict__ hsrc,
                               const int* __restrict__ src,
                               const int* __restrict__ dst,
                               float* __restrict__ agg,
                               float* __restrict__ deg, int E) {
    int tid = blockIdx.x * blockDim.x + threadIdx.x;
    int e = tid >> 5, lane = tid & 31;
    if (e >= E) return;
    int s = src[e], d = dst[e];
    const float4* row = (const float4*)(hsrc + (size_t)s * H);
    float4 v = row[lane];
    float* ap = agg + (size_t)d * H + lane * 4;
    unsafeAtomicAdd(ap + 0, v.x);
    unsafeAtomicAdd(ap + 1, v.y);
    unsafeAtomicAdd(ap + 2, v.z);
    unsafeAtomicAdd(ap + 3, v.w);
    if (lane == 0) unsafeAtomicAdd(deg + d, 1.0f);
}

__global__ void normalize_kernel(float* __restrict__ agg,
                                 const float* __restrict__ deg, size_t n) {
    size_t i = (size_t)blockIdx.x * blockDim.x + threadIdx.x;
    if (i >= n) return;
    agg[i] = agg[i] / fmaxf(deg[i >> 7], 1.0f);  // H == 128 -> shift 7
}

__global__ void bias_act_kernel(float* __restrict__ h, const float* __restrict__ bias,
                                size_t n, int dout, int act) {
    size_t i = (size_t)blockIdx.x * blockDim.x + threadIdx.x;
    if (i >= n) return;
    int o = (int)(i % (size_t)dout);
    float v = h[i] + bias[o];
    h[i] = act ? fmaxf(v, 0.0f) : v;
}

// ---------------- WMMA GEMM: C[M x dout] += A[M x 128] @ W[128 x dout] ----------------
// Wt is W transposed: Wt[n][k], n = output column.
// block = (32, dout/16): each wave owns one 16-column N-tile of a 16-row M-tile.
// fp32 WMMA (V_WMMA_F32_16X16X4_F32): exact fp32, matches reference numerics.
// M is guaranteed to be a multiple of 16 (50000 / 100000) -> no bounds checks.
__global__ void gemm_wmma_kernel(const float* __restrict__ A,
                                 const float* __restrict__ Wt,
                                 float* __restrict__ C, int dout) {
    __shared__ float As[16 * ASTR];
    const int lane = threadIdx.x;           // 0..31 (wave32)
    const int nt   = threadIdx.y;           // N-tile index within block
    const int nthreads = blockDim.y * 32;
    const int tflat = nt * 32 + lane;
    const int bm = blockIdx.x;

    // Stage the 16x128 A tile straight into LDS via the async global->LDS path
    // (GLOBAL_LOAD_ASYNC_TO_LDS_B128, tracked by ASYNCcnt; 16B per lane per issue).
    // The LDS operand MUST be derived from &As[...] (ptrtoint escape) so the
    // compiler knows the asm writes As; the low 32 bits of a flat shared address
    // are the wave-relative LDS byte address the instruction's VDST expects.
    for (int i = tflat * 4; i < 16 * H; i += nthreads * 4) {
        int row = i >> 7, col = i & (H - 1);
        unsigned ldsOff = (unsigned)(size_t)(&As[row * ASTR + col]);
        const float* gp = A + (size_t)(bm * 16 + row) * H + col;
        asm volatile("global_load_async_to_lds_b128 %0, %1, off"
                     :: "v"(ldsOff), "v"(gp) : "memory");
    }
    asm volatile("s_wait_asynccnt 0" ::: "memory");
    __syncthreads();

    const int half = lane >> 4;       // 0: lanes 0-15, 1: lanes 16-31
    const int m    = lane & 15;
    const int n    = nt * 16 + m;     // this lane's output column
    const float* arow = &As[m * ASTR];
    const float* brow = Wt + (size_t)n * H;

    v8f c = {0.f, 0.f, 0.f, 0.f, 0.f, 0.f, 0.f, 0.f};
    for (int k0 = 0; k0 < H; k0 += 4) {
        int k = k0 + half * 2;        // ISA A/B layout: vgpr j -> K = j + 2*half
        v2f a = *(const v2f*)(arow + k);   // ds_load_b64
        v2f b = *(const v2f*)(brow + k);   // global_load_b64 (L2-resident W)
        c = __builtin_amdgcn_wmma_f32_16x16x4_f32(false, a, false, b,
                                                  (short)0, c, false, false);
    }

    // C/D layout: vgpr v -> row (v + 8*half), col n. Accumulate with hardware fp32
    // atomics: one global_atomic_add_f32 per element, no RMW dependency chain.
    float* crow = C + (size_t)(bm * 16 + half * 8) * dout + n;
#pragma unroll
    for (int v = 0; v < 8; ++v) {
        unsafeAtomicAdd(crow + (size_t)v * dout, c[v]);
    }
}

// ---------------- host orchestration ----------------

extern "C" void kernel_launch(void* const* d_in, const int* in_sizes, int n_in,
                              void* d_out, int out_size, void* d_ws, size_t ws_size,
                              hipStream_t stream) {
    const float* feat_a = (const float*)d_in[0];
    const float* feat_b = (const float*)d_in[1];
    const float* feat_d = (const float*)d_in[2];
    const float* basis[3] = {(const float*)d_in[3], (const float*)d_in[6], (const float*)d_in[9]};
    const float* coef[3]  = {(const float*)d_in[4], (const float*)d_in[7], (const float*)d_in[10]};
    const float* bias[3]  = {(const float*)d_in[5], (const float*)d_in[8], (const float*)d_in[11]};
    const int* esrc[4] = {(const int*)d_in[12], (const int*)d_in[14], (const int*)d_in[16], (const int*)d_in[18]};
    const int* edst[4] = {(const int*)d_in[13], (const int*)d_in[15], (const int*)d_in[17], (const int*)d_in[19]};
    const int E = in_sizes[12];

    float* ws = (float*)d_ws;
    size_t o = 0;
    float* hA0 = ws + o; o += (size_t)NAx * H;
    float* hB0 = ws + o; o += (size_t)NBx * H;
    float* hD0 = ws + o; o += (size_t)NDx * H;
    float* hA1 = ws + o; o += (size_t)NAx * H;
    float* hB1 = ws + o; o += (size_t)NBx * H;
    float* hD1 = ws + o; o += (size_t)NDx * H;
    float* agg = ws + o; o += (size_t)NAx * H;   // reused per relation (max dst count)
    float* deg = ws + o; o += (size_t)NAx;
    float* Wt  = ws + o; o += (size_t)4 * H * H;
    (void)ws_size; (void)n_in; (void)out_size;

    auto zero = [&](float* p, size_t n) {
        int blocks = (int)((n + 256 * 8 - 1) / (256 * 8));
        if (blocks > 8192) blocks = 8192;
        if (blocks < 1) blocks = 1;
        zero_kernel<<<blocks, 256, 0, stream>>>(p, n);
    };

    auto layer = [&](const float* ha, const float* hb, const float* hd, int l,
                     int dout, bool act, bool only_d,
                     float* oa, float* ob, float* od) {
        int wtot = 4 * H * dout;
        compute_Wt_kernel<<<(wtot + 255) / 256, 256, 0, stream>>>(basis[l], coef[l], Wt, dout);
        if (!only_d) {
            zero(oa, (size_t)NAx * dout);
            zero(ob, (size_t)NBx * dout);
        }
        zero(od, (size_t)NDx * dout);

        const float* srcH[4] = {ha, hb, hd, hd};   // RELS: a->d, b->d, d->a, d->b
        int    ndst[4] = {NDx, NDx, NAx, NBx};
        float* outp[4] = {od, od, oa, ob};
        int nrel = only_d ? 2 : 4;                 // last layer only needs dst 'd'

        for (int r = 0; r < nrel; ++r) {
            int nd = ndst[r];
            zero(agg, (size_t)nd * H);
            zero(deg, (size_t)nd);
            int sthreads = E * 32;
            scatter_kernel<<<(sthreads + 255) / 256, 256, 0, stream>>>(
                srcH[r], esrc[r], edst[r], agg, deg, E);
            size_t ntot = (size_t)nd * H;
            normalize_kernel<<<(int)((ntot + 255) / 256), 256, 0, stream>>>(agg, deg, ntot);
            dim3 blk(32, dout / 16, 1);
            gemm_wmma_kernel<<<dim3(nd / 16), blk, 0, stream>>>(
                agg, Wt + (size_t)r * H * dout, outp[r], dout);
        }

        int actI = act ? 1 : 0;
        if (!only_d) {
            size_t na = (size_t)NAx * dout;
            bias_act_kernel<<<(int)((na + 255) / 256), 256, 0, stream>>>(oa, bias[l], na, dout, actI);
            size_t nb = (size_t)NBx * dout;
            bias_act_kernel<<<(int)((nb + 255) / 256), 256, 0, stream>>>(ob, bias[l], nb, dout, actI);
        }
        size_t ndd = (size_t)NDx * dout;
        bias_act_kernel<<<(int)((ndd + 255) / 256), 256, 0, stream>>>(od, bias[l], ndd, dout, actI);
    };

    layer(feat_a, feat_b, feat_d, 0, H,    true,  false, hA0, hB0, hD0);
    layer(hA0,    hB0,    hD0,    1, H,    true,  false, hA1, hB1, hD1);
    layer(hA1,    hB1,    hD1,    2, OUTD, false, true,  nullptr, nullptr, (float*)d_out);
}